// LorentzBatchNorm_beta_21199958573182
// MI455X (gfx1250) — compile-verified
//
#include <hip/hip_runtime.h>
#include <hip/hip_bf16.h>

#define DD 256
#define BB 128
#define SS 1024
#define ROWS (BB*SS)
#define NBLK3 512
#define NBLK4 2048

typedef float v2f __attribute__((ext_vector_type(2)));
typedef float v8f __attribute__((ext_vector_type(8)));

__device__ __forceinline__ float wred32(float v) {
#pragma unroll
  for (int off = 16; off; off >>= 1) v += __shfl_xor(v, off, 32);
  return v;
}

// ---------------------------------------------------------------------------
// K1: per-batch centroid over S.  Sum over S done as ones[16x4] x X[4x16]
// with V_WMMA_F32_16X16X4_F32 (A = all ones -> every row of D is the column
// sum; only the N = lane%16 column striping of B matters).
// grid = BB blocks, 512 threads (16 waves; wave w owns features [16w,16w+16)).
// ---------------------------------------------------------------------------
__global__ void k1_centroid_seq(const float* __restrict__ x, float* __restrict__ cb) {
  const int b    = blockIdx.x;
  const int tid  = threadIdx.x;
  const int wave = tid >> 5;
  const int lane = tid & 31;
  const int col  = lane & 15;
  const int half = lane >> 4;
  const float* xb = x + (size_t)b * SS * DD;
  __shared__ float s_sum[DD];
  __shared__ float s_red[DD];

#if __has_builtin(__builtin_amdgcn_wmma_f32_16x16x4_f32)
  {
    v8f c = {0.f,0.f,0.f,0.f,0.f,0.f,0.f,0.f};
    v2f a; a[0] = 1.0f; a[1] = 1.0f;
    const int fbase = wave * 16 + col;
    const int k0 = half * 2;
    for (int s0 = 0; s0 < SS; s0 += 4) {
      v2f bm;
      bm[0] = xb[(size_t)(s0 + k0    ) * DD + fbase];
      bm[1] = xb[(size_t)(s0 + k0 + 1) * DD + fbase];
      // D = A(ones,16x4) * B(4x16) + C : accumulate column sums over S
      c = __builtin_amdgcn_wmma_f32_16x16x4_f32(false, a, false, bm,
                                                (short)0, c, false, false);
    }
    if (lane < 16) s_sum[wave * 16 + col] = c[0];   // D row M=0 = column sums
  }
#else
  if (tid < DD) {
    float acc = 0.f;
    for (int s = 0; s < SS; ++s) acc += xb[(size_t)s * DD + tid];
    s_sum[tid] = acc;
  }
#endif
  __syncthreads();
  if (tid < DD) {
    float a2 = s_sum[tid] * (1.0f / SS);
    s_sum[tid] = a2;
    float sq = a2 * a2;
    s_red[tid] = (tid == 0) ? -sq : sq;   // Minkowski <avg,avg>
  }
  __syncthreads();
  for (int str = 128; str >= 1; str >>= 1) {
    if (tid < str) s_red[tid] += s_red[tid + str];
    __syncthreads();
  }
  if (tid < DD) {
    float denom = sqrtf(fmaxf(fabsf(s_red[0]), 1e-8f));
    cb[b * DD + tid] = s_sum[tid] / denom;
  }
}

// ---------------------------------------------------------------------------
// K2: centroid over batch dim.  grid=1, block=256.
// ---------------------------------------------------------------------------
__global__ void k2_centroid_batch(const float* __restrict__ cb, float* __restrict__ mean) {
  const int t = threadIdx.x;
  __shared__ float s_red[DD];
  float acc = 0.f;
  for (int b = 0; b < BB; ++b) acc += cb[b * DD + t];
  float a = acc * (1.0f / BB);
  float sq = a * a;
  s_red[t] = (t == 0) ? -sq : sq;
  __syncthreads();
  for (int str = 128; str >= 1; str >>= 1) {
    if (t < str) s_red[t] += s_red[t + str];
    __syncthreads();
  }
  float denom = sqrtf(fmaxf(fabsf(s_red[0]), 1e-8f));
  mean[t] = a / denom;
}

// ---------------------------------------------------------------------------
// K3: per-feature sum / sumsq of tangent vectors (logmap + transp0back).
// Wave-per-row, lane owns 8 contiguous features, accumulators in VGPRs,
// block combine via LDS atomics, per-block partials to workspace.
// grid = NBLK3, block = 256 (8 waves).
// ---------------------------------------------------------------------------
__global__ void k3_stats(const float* __restrict__ x, const float* __restrict__ mean,
                         float* __restrict__ part) {
  const int tid  = threadIdx.x;
  const int lane = tid & 31;
  const int wave = tid >> 5;
  __shared__ float ls[DD], lq[DD];
  ls[tid] = 0.f; lq[tid] = 0.f;
  __syncthreads();

  float m[8];
  {
    const float4* mp = (const float4*)mean;
    float4 a = mp[lane * 2], b = mp[lane * 2 + 1];
    m[0]=a.x; m[1]=a.y; m[2]=a.z; m[3]=a.w; m[4]=b.x; m[5]=b.y; m[6]=b.z; m[7]=b.w;
  }
  const float m0   = __shfl(m[0], 0, 32);
  const float invm = 1.0f / (1.0f + m0);

  float as[8], aq[8];
#pragma unroll
  for (int j = 0; j < 8; ++j) { as[j] = 0.f; aq[j] = 0.f; }

  const int gw = blockIdx.x * 8 + wave;
  const int nw = NBLK3 * 8;
  for (int r = gw; r < ROWS; r += nw) {
    const float4* row = (const float4*)(x + (size_t)r * DD);
    float4 x0 = row[lane * 2], x1 = row[lane * 2 + 1];
    float xr[8] = {x0.x,x0.y,x0.z,x0.w,x1.x,x1.y,x1.z,x1.w};
    float p = 0.f;
#pragma unroll
    for (int j = 0; j < 8; ++j) p += m[j] * xr[j];
    if (lane == 0) p -= 2.0f * m[0] * xr[0];        // Minkowski sign on d=0
    const float inner = wred32(p);
    const float alpha = fmaxf(-inner, 1.0f + 1e-7f);
    const float f = acoshf(alpha) / sqrtf(alpha * alpha - 1.0f);
    float tl[8];
#pragma unroll
    for (int j = 0; j < 8; ++j) tl[j] = f * (xr[j] - alpha * m[j]);
    const float t0   = __shfl(tl[0], 0, 32);
    const float coef = t0 * invm;
#pragma unroll
    for (int j = 0; j < 8; ++j) {
      float o = (j == 0 && lane == 0) ? 1.0f : 0.0f;
      float v = tl[j] - coef * (m[j] + o);
      as[j] += v; aq[j] += v * v;
    }
  }
#pragma unroll
  for (int j = 0; j < 8; ++j) {
    atomicAdd(&ls[lane * 8 + j], as[j]);   // ds_add_f32
    atomicAdd(&lq[lane * 8 + j], aq[j]);
  }
  __syncthreads();
  part[(size_t)blockIdx.x * 512 + tid]      = ls[tid];
  part[(size_t)blockIdx.x * 512 + DD + tid] = lq[tid];
}

// ---------------------------------------------------------------------------
// K3b: reduce partials -> scale[d] = gamma / sqrt(var + 1e-5). grid=1, 512 thr.
// ---------------------------------------------------------------------------
__global__ void k3b_scale(const float* __restrict__ part, const float* __restrict__ gamma,
                          float* __restrict__ scale) {
  const int t = threadIdx.x;
  __shared__ float tot[512];
  float acc = 0.f;
  for (int i = 0; i < NBLK3; ++i) acc += part[(size_t)i * 512 + t];
  tot[t] = acc;
  __syncthreads();
  if (t < DD) {
    const float N = (float)ROWS;
    float mn  = tot[t] / N;
    float var = tot[t + DD] / N - mn * mn;
    scale[t] = gamma[0] / sqrtf(var + 1e-5f);
  }
}

// ---------------------------------------------------------------------------
// K4: recompute tangent (x is L2-resident after K1/K3 on 192MB L2), rescale,
// transp0(beta), expmap(beta) -> out.  grid = NBLK4, block = 256.
// ---------------------------------------------------------------------------
__global__ void k4_out(const float* __restrict__ x, const float* __restrict__ mean,
                       const float* __restrict__ beta, const float* __restrict__ scale,
                       float* __restrict__ out) {
  const int tid  = threadIdx.x;
  const int lane = tid & 31;
  const int wave = tid >> 5;
  float m[8], bt[8], sc[8];
  {
    const float4* mp = (const float4*)mean;
    const float4* bp = (const float4*)beta;
    const float4* sp = (const float4*)scale;
    float4 a, b;
    a = mp[lane*2]; b = mp[lane*2+1];
    m[0]=a.x; m[1]=a.y; m[2]=a.z; m[3]=a.w; m[4]=b.x; m[5]=b.y; m[6]=b.z; m[7]=b.w;
    a = bp[lane*2]; b = bp[lane*2+1];
    bt[0]=a.x; bt[1]=a.y; bt[2]=a.z; bt[3]=a.w; bt[4]=b.x; bt[5]=b.y; bt[6]=b.z; bt[7]=b.w;
    a = sp[lane*2]; b = sp[lane*2+1];
    sc[0]=a.x; sc[1]=a.y; sc[2]=a.z; sc[3]=a.w; sc[4]=b.x; sc[5]=b.y; sc[6]=b.z; sc[7]=b.w;
  }
  const float m0   = __shfl(m[0], 0, 32);
  const float invm = 1.0f / (1.0f + m0);
  const float b0   = __shfl(bt[0], 0, 32);
  const float invb = 1.0f / (1.0f + b0);

  const int gw = blockIdx.x * 8 + wave;
  const int nw = NBLK4 * 8;
  for (int r = gw; r < ROWS; r += nw) {
    const float4* row = (const float4*)(x + (size_t)r * DD);
    float4 x0 = row[lane * 2], x1 = row[lane * 2 + 1];
    float xr[8] = {x0.x,x0.y,x0.z,x0.w,x1.x,x1.y,x1.z,x1.w};
    float p = 0.f;
#pragma unroll
    for (int j = 0; j < 8; ++j) p += m[j] * xr[j];
    if (lane == 0) p -= 2.0f * m[0] * xr[0];
    const float inner = wred32(p);
    const float alpha = fmaxf(-inner, 1.0f + 1e-7f);
    const float f = acoshf(alpha) / sqrtf(alpha * alpha - 1.0f);
    float tl[8];
#pragma unroll
    for (int j = 0; j < 8; ++j) tl[j] = f * (xr[j] - alpha * m[j]);
    const float t0   = __shfl(tl[0], 0, 32);
    const float coef = t0 * invm;
    float w[8];
#pragma unroll
    for (int j = 0; j < 8; ++j) {
      float o = (j == 0 && lane == 0) ? 1.0f : 0.0f;
      w[j] = (tl[j] - coef * (m[j] + o)) * sc[j];
    }
    // transp0(beta, w)
    float p2 = 0.f;
#pragma unroll
    for (int j = 0; j < 8; ++j) p2 += bt[j] * w[j];
    if (lane == 0) p2 -= 2.0f * bt[0] * w[0];
    const float coef2 = wred32(p2) * invb;
    float y[8];
#pragma unroll
    for (int j = 0; j < 8; ++j) {
      float o = (j == 0 && lane == 0) ? 1.0f : 0.0f;
      y[j] = w[j] + coef2 * (bt[j] + o);
    }
    // expmap(beta, y)
    float p3 = 0.f;
#pragma unroll
    for (int j = 0; j < 8; ++j) p3 += y[j] * y[j];
    if (lane == 0) p3 -= 2.0f * y[0] * y[0];
    float nn = fmaxf(wred32(p3), 1e-7f);
    float n  = sqrtf(nn);
    float ch = coshf(n);
    float sn = sinhf(n) / n;
    float4* orow = (float4*)(out + (size_t)r * DD);
    float4 o0, o1;
    o0.x = ch*bt[0] + sn*y[0]; o0.y = ch*bt[1] + sn*y[1];
    o0.z = ch*bt[2] + sn*y[2]; o0.w = ch*bt[3] + sn*y[3];
    o1.x = ch*bt[4] + sn*y[4]; o1.y = ch*bt[5] + sn*y[5];
    o1.z = ch*bt[6] + sn*y[6]; o1.w = ch*bt[7] + sn*y[7];
    orow[lane * 2]     = o0;
    orow[lane * 2 + 1] = o1;
  }
}

extern "C" void kernel_launch(void* const* d_in, const int* in_sizes, int n_in,
                              void* d_out, int out_size, void* d_ws, size_t ws_size,
                              hipStream_t stream) {
  const float* x     = (const float*)d_in[0];
  const float* beta  = (const float*)d_in[1];
  const float* gamma = (const float*)d_in[2];
  float* out = (float*)d_out;
  float* ws  = (float*)d_ws;
  float* cb    = ws;                  // BB*DD
  float* mean  = cb + BB * DD;        // DD
  float* scale = mean + DD;           // DD
  float* part  = scale + DD;          // NBLK3*512

  k1_centroid_seq<<<BB, 512, 0, stream>>>(x, cb);
  k2_centroid_batch<<<1, DD, 0, stream>>>(cb, mean);
  k3_stats<<<NBLK3, 256, 0, stream>>>(x, mean, part);
  k3b_scale<<<1, 512, 0, stream>>>(part, gamma, scale);
  k4_out<<<NBLK4, 256, 0, stream>>>(x, mean, beta, scale, out);
}